// RelationshipAwareModule_83734682403555
// MI455X (gfx1250) — compile-verified
//
#include <hip/hip_runtime.h>

#define BN_EPS 1e-5f

typedef __attribute__((ext_vector_type(2))) float v2f;
typedef __attribute__((ext_vector_type(8))) float v8f;

// ---------------------------------------------------------------------------
// Kernel 1: conv3 (10->1, 3x3, SAME) + bias + BatchNorm(eval) -> y (64,32,32)
// ---------------------------------------------------------------------------
__global__ void conv_bn_kernel(const float* __restrict__ x,      // (64,10,32,32)
                               const float* __restrict__ w,      // (1,10,3,3)
                               const float* __restrict__ cb,     // (1,)
                               const float* __restrict__ gamma,
                               const float* __restrict__ beta,
                               const float* __restrict__ mean,
                               const float* __restrict__ var,
                               float* __restrict__ y) {          // (64,32,32)
  __shared__ float ws[90];
  const int t = threadIdx.x;
  const int b = blockIdx.x;
  if (t < 90) ws[t] = w[t];
  __syncthreads();
  const float scale = gamma[0] * rsqrtf(var[0] + BN_EPS);
  const float shift = beta[0] - mean[0] * scale;
  const float bia = cb[0];
  const float* xb = x + b * 10 * 1024;
#pragma unroll
  for (int q = 0; q < 4; ++q) {
    const int p = t + q * 256;
    const int r = p >> 5, c = p & 31;
    float acc = 0.f;
    for (int ch = 0; ch < 10; ++ch) {
      const float* xc = xb + ch * 1024;
      const float* wc = ws + ch * 9;
#pragma unroll
      for (int kh = 0; kh < 3; ++kh) {
        const int rr = r + kh - 1;
        if (rr < 0 || rr > 31) continue;
#pragma unroll
        for (int kw = 0; kw < 3; ++kw) {
          const int cc = c + kw - 1;
          if (cc < 0 || cc > 31) continue;
          acc += xc[rr * 32 + cc] * wc[kh * 3 + kw];
        }
      }
    }
    y[b * 1024 + p] = (acc + bia) * scale + shift;
  }
}

// ---------------------------------------------------------------------------
// Kernel 2: build A (640 x 2048): A[b*10+c, 2p+0] = x[b,c,p], A[.,2p+1] = y[b,p]
// (interleave once so the WMMA K-loop reads contiguous float2 per lane)
// ---------------------------------------------------------------------------
__global__ void build_A_kernel(const float* __restrict__ x,
                               const float* __restrict__ y,
                               float* __restrict__ A) {
  const int r = blockIdx.x;            // 0..639  (= b*10 + c)
  const int b = r / 10;
  const int t = threadIdx.x;
  const float* xp = x + r * 1024;
  const float* yp = y + b * 1024;
  float2* Ap = (float2*)(A + r * 2048);
#pragma unroll
  for (int q = 0; q < 4; ++q) {
    const int p = t + q * 256;
    Ap[p] = make_float2(xp[p], yp[p]);
  }
}

// ---------------------------------------------------------------------------
// Kernel 3: h = A(640x2048) @ fc1_w^T(2048x128) + fc1_b  via V_WMMA_F32_16X16X4_F32
// One wave per 16x16 C tile; 40x8 tiles = 320 waves = 40 blocks of 256 threads.
// A fragment (16x4 f32, 2 VGPRs): lane l(0..15) -> M=l; v0=K(0|2), v1=K(1|3)
// B fragment (4x16 f32, 2 VGPRs): lane l(0..15) -> N=l; v0=K(0|2), v1=K(1|3)
// Both are contiguous 8B loads per lane per K-step.
// ---------------------------------------------------------------------------
__global__ void fc1_wmma_kernel(const float* __restrict__ A,
                                const float* __restrict__ W,    // (128,2048)
                                const float* __restrict__ bias, // (128,)
                                float* __restrict__ H) {        // (640,128)
  const int wave  = (blockIdx.x * blockDim.x + threadIdx.x) >> 5;
  const int lane  = threadIdx.x & 31;
  const int tileM = wave >> 3;          // 0..39
  const int tileN = wave & 7;           // 0..7
  const int half  = lane >> 4;          // 0 | 1
  const int l     = lane & 15;
  const float* arow = A + (tileM * 16 + l) * 2048 + 2 * half;
  const float* brow = W + (tileN * 16 + l) * 2048 + 2 * half;
  v8f acc = {};
  for (int k = 0; k < 2048; k += 4) {
    v2f a = *(const v2f*)(arow + k);
    v2f b = *(const v2f*)(brow + k);
    acc = __builtin_amdgcn_wmma_f32_16x16x4_f32(false, a, false, b,
                                                (short)0, acc, false, false);
  }
  // C/D layout: VGPR r holds M = tileM*16 + half*8 + r, N = tileN*16 + l
  const int row0 = tileM * 16 + half * 8;
  const int col  = tileN * 16 + l;
  const float bb = bias[col];
#pragma unroll
  for (int r = 0; r < 8; ++r) {
    H[(row0 + r) * 128 + col] = acc[r] + bb;
  }
}

// fast deterministic tanh via v_exp_f32: tanh(v) = sign(v)*(1 - 2/(e^{2|v|}+1))
__device__ __forceinline__ float fast_tanh(float v) {
  const float av = fabsf(v);
  const float t = 1.0f - 2.0f / (__expf(2.0f * av) + 1.0f);
  return copysignf(t, v);
}

// ---------------------------------------------------------------------------
// Kernel 4: fused attention: k -> tanh -> softmax(axis=e) -> gate by x ->
//           3x3/stride-3 mean pool -> 10x10 FC.  One block per (b,c,a).
//           Deterministic: LDS staging + fixed-order reductions (no atomics).
// ---------------------------------------------------------------------------
__global__ __launch_bounds__(1024)
void attention_kernel(const float* __restrict__ x,     // (64,10,32,32)
                      const float* __restrict__ H,     // (640,128)
                      const float* __restrict__ awc,   // (32,)
                      const float* __restrict__ abc,   // (20,32,32)
                      const float* __restrict__ awi,   // (20,32,32)
                      const float* __restrict__ abi,   // (20,32,32)
                      const float* __restrict__ fcw,   // (10,10)
                      const float* __restrict__ fcb,   // (10,)
                      float* __restrict__ out) {       // (20,64,10,4,10,10)
  __shared__ float g_s[20 * 1024];   // gated maps, 80 KB
  __shared__ float p_s[2000];        // pooled (20,10,10)
  __shared__ float fcw_s[100];
  __shared__ float fcb_s[10];

  const int t   = threadIdx.x;       // 0..1023 == spatial position
  const int blk = blockIdx.x;        // ((b*10+c)*4 + a)
  const int a   = blk & 3;
  const int bc  = blk >> 2;          // b*10 + c
  const int b   = bc / 10;
  const int c   = bc - b * 10;

  if (t < 100) fcw_s[t] = fcw[t];
  else if (t < 110) fcb_s[t - 100] = fcb[t - 100];

  // k[b,c,a,r,cc] = awc[a*8 + pos>>7] * h[bc, pos&127]
  const int pos  = t;
  const float kv = awc[a * 8 + (pos >> 7)] * H[bc * 128 + (pos & 127)];
  const float xv = x[bc * 1024 + pos];

  float logit[20];
  float mx = -3.4e38f;
#pragma unroll
  for (int e = 0; e < 20; ++e) {
    const int idx = e * 1024 + pos;
    const float z = fast_tanh(kv + abc[idx]);
    const float l = awi[idx] * z + abi[idx];
    logit[e] = l;
    mx = fmaxf(mx, l);
  }
  float sum = 0.f;
#pragma unroll
  for (int e = 0; e < 20; ++e) {
    const float ex = __expf(logit[e] - mx);
    logit[e] = ex;
    sum += ex;
  }
  const float gscale = xv / sum;
#pragma unroll
  for (int e = 0; e < 20; ++e) g_s[e * 1024 + pos] = logit[e] * gscale;
  __syncthreads();

  // pool: p[e,pr,pc] = mean over 3x3 of g[e, 3pr+dr, 3pc+dc]
  if (t < 2000) {
    const int e = t / 100;
    const int rem = t - e * 100;
    const int pr = rem / 10;
    const int pc = rem - pr * 10;
    const float* gp = &g_s[e * 1024 + (pr * 3) * 32 + pc * 3];
    float s = 0.f;
#pragma unroll
    for (int dr = 0; dr < 3; ++dr)
#pragma unroll
      for (int dc = 0; dc < 3; ++dc) s += gp[dr * 32 + dc];
    p_s[t] = s * (1.0f / 9.0f);
  }
  __syncthreads();

  // out[e,b,c,a,pr,n] = sum_pc p[e,pr,pc] * fcw[n,pc] + fcb[n]
  if (t < 2000) {
    const int e = t / 100;
    const int rem = t - e * 100;
    const int pr = rem / 10;
    const int n = rem - pr * 10;
    const float* pp = &p_s[e * 100 + pr * 10];
    float acc = fcb_s[n];
#pragma unroll
    for (int pc = 0; pc < 10; ++pc) acc += pp[pc] * fcw_s[n * 10 + pc];
    out[((((e * 64 + b) * 10 + c) * 4 + a) * 10 + pr) * 10 + n] = acc;
  }
}

// ---------------------------------------------------------------------------
extern "C" void kernel_launch(void* const* d_in, const int* in_sizes, int n_in,
                              void* d_out, int out_size, void* d_ws, size_t ws_size,
                              hipStream_t stream) {
  const float* x      = (const float*)d_in[0];
  const float* conv_w = (const float*)d_in[1];
  const float* conv_b = (const float*)d_in[2];
  const float* gamma  = (const float*)d_in[3];
  const float* beta   = (const float*)d_in[4];
  const float* mean   = (const float*)d_in[5];
  const float* var    = (const float*)d_in[6];
  const float* fc1_w  = (const float*)d_in[7];
  const float* fc1_b  = (const float*)d_in[8];
  const float* awc    = (const float*)d_in[9];
  const float* abc    = (const float*)d_in[10];
  const float* awi    = (const float*)d_in[11];
  const float* abi    = (const float*)d_in[12];
  const float* fcw    = (const float*)d_in[13];
  const float* fcb    = (const float*)d_in[14];
  float* out = (float*)d_out;

  // workspace: y (64*1024) | A (640*2048) | H (640*128)
  float* y = (float*)d_ws;
  float* A = y + 64 * 1024;
  float* H = A + 640 * 2048;

  conv_bn_kernel<<<64, 256, 0, stream>>>(x, conv_w, conv_b, gamma, beta, mean, var, y);
  build_A_kernel<<<640, 256, 0, stream>>>(x, y, A);
  fc1_wmma_kernel<<<40, 256, 0, stream>>>(A, fc1_w, fc1_b, H);
  attention_kernel<<<2560, 1024, 0, stream>>>(x, H, awc, abc, awi, abi, fcw, fcb, out);
}